// MessagePassingEncoder_58789512347872
// MI455X (gfx1250) — compile-verified
//
#include <hip/hip_runtime.h>
#include <hip/hip_bf16.h>
#include <math.h>

// Problem constants (from reference): B=8, N=200, D=128, L=3, H1=64, H2=32.
#define NB   8
#define NN   200
#define NP   208      // N padded to multiple of 16 (13 tiles)
#define DD   128
#define H1   64
#define H2   32
#define NLAY 3
#define LSTR 68       // padded LDS row stride (floats): banks = 4*row + col,
                      // conflict-free b64 loads across both half-waves

typedef __attribute__((ext_vector_type(2))) float v2f;
typedef __attribute__((ext_vector_type(8))) float v8f;

__device__ __forceinline__ float relu_(float x) { return x > 0.f ? x : 0.f; }

// ---------------------------------------------------------------------------
// init_h = locs @ W_init + b_init  -> padded nf (pad rows zeroed) and d_out[1]
// ---------------------------------------------------------------------------
__global__ void init_kernel(const float* __restrict__ locs,
                            const float* __restrict__ Wi,
                            const float* __restrict__ bi,
                            float* __restrict__ nf,
                            float* __restrict__ out_init) {
  int t = blockIdx.x * blockDim.x + threadIdx.x;
  if (t >= NB * NP * DD) return;
  int d = t % DD;
  int i = (t / DD) % NP;
  int b = t / (DD * NP);
  float v = 0.f;
  if (i < NN) {
    float l0 = locs[(b * NN + i) * 2 + 0];
    float l1 = locs[(b * NN + i) * 2 + 1];
    v = fmaf(l0, Wi[d], fmaf(l1, Wi[DD + d], bi[d]));
    out_init[(b * NN + i) * DD + d] = v;
  }
  nf[t] = v;
}

// ---------------------------------------------------------------------------
// ef[b,r,c] = || init_h[b,r,:] - init_h[b,c,:] ||  (pad entries -> 0)
// ---------------------------------------------------------------------------
__global__ __launch_bounds__(256) void ef_kernel(const float* __restrict__ nf,
                                                 float* __restrict__ ef) {
  __shared__ float rA[16][DD + 2];
  __shared__ float rB[16][DD + 2];
  int b = blockIdx.z;
  int rt = blockIdx.y * 16, ct = blockIdx.x * 16;
  int tid = threadIdx.x;
  for (int i = tid; i < 16 * DD; i += 256) {
    int rr = i >> 7, d = i & 127;
    rA[rr][d] = nf[((b * NP) + (rt + rr)) * DD + d];
    rB[rr][d] = nf[((b * NP) + (ct + rr)) * DD + d];
  }
  __syncthreads();
  int lc = tid & 15, lr = tid >> 4;
  int r = rt + lr, c = ct + lc;
  float s = 0.f;
#pragma unroll 8
  for (int d = 0; d < DD; ++d) {
    float df = rA[lr][d] - rB[lc][d];
    s = fmaf(df, df, s);
  }
  float v = (r < NN && c < NN) ? sqrtf(s) : 0.f;
  ef[((size_t)(b * NP) + r) * NP + c] = v;
}

// ---------------------------------------------------------------------------
// Separable factorization of the 257->64 edge layer:
//   P = nf @ ew0[l][0:128,:],  Q = nf @ ew0[l][128:256,:]
// ---------------------------------------------------------------------------
__global__ __launch_bounds__(64) void pq_kernel(const float* __restrict__ nf,
                                                const float* __restrict__ w0,
                                                float* __restrict__ P,
                                                float* __restrict__ Q) {
  __shared__ float x[DD];
  int bi = blockIdx.x;      // 0 .. NB*NP-1
  int t = threadIdx.x;      // 0 .. 63
  x[t] = nf[bi * DD + t];
  x[t + 64] = nf[bi * DD + t + 64];
  __syncthreads();
  float p = 0.f, q = 0.f;
#pragma unroll 8
  for (int d = 0; d < DD; ++d) {
    float xv = x[d];
    p = fmaf(xv, w0[d * H1 + t], p);
    q = fmaf(xv, w0[(128 + d) * H1 + t], q);
  }
  P[bi * H1 + t] = p;
  Q[bi * H1 + t] = q;
}

// ---------------------------------------------------------------------------
// Edge MLP core (dominant compute) using exact-fp32 CDNA5 WMMA 16x16x4.
//   h0[m][k]  = relu((P[r][k]+b0[k]) + Q[c][k] + ef*w0e[k])   (A built in regs)
//   h1        = relu(h0 @ W1 + b1)                            (32 v_wmma ops)
//   uef[m]    = h1 . w2 + b2                                  (shfl_xor dot)
// One wave = 16 edges (one r, 16 cols) x 32 hidden; block = 8 waves = 8 rows.
// A layout (ISA 16x4 f32): lane L holds M=L%16, K = kb,kb+1, kb = 4*kk+2*(L/16).
// B mirrored: lane L holds N=L%16, K rows kb,kb+1 -> contiguous in sW1T.
// ---------------------------------------------------------------------------
__global__ __launch_bounds__(256) void edge_mlp_kernel(
    const float* __restrict__ P, const float* __restrict__ Q,
    const float* __restrict__ efIn, float* __restrict__ uefOut,
    const float* __restrict__ w0, const float* __restrict__ b0,
    const float* __restrict__ w1, const float* __restrict__ b1,
    const float* __restrict__ w2, const float* __restrict__ b2) {
  __shared__ float sW1T[H2][LSTR];   // transposed W1: sW1T[n][k] = W1[k][n]
  __shared__ float sPb[8][H1];       // P row + b0 (bias folded)
  __shared__ float sQ[16][LSTR];     // Q rows, padded stride
  __shared__ float sW0e[H1];
  __shared__ float sB1[H2];
  __shared__ float sW2[H2];
  __shared__ float sEf[8][16];

  const int b = blockIdx.z;
  const int r0 = blockIdx.y * 8;
  const int c0 = blockIdx.x * 16;
  const int tid = threadIdx.x;

  for (int i = tid; i < H2 * H1; i += 256) {
    int n = i >> 6, k = i & 63;               // 32 x 64
    sW1T[n][k] = w1[k * H2 + n];
  }
  for (int i = tid; i < 8 * H1; i += 256) {
    int rr = i >> 6, kk = i & 63;
    sPb[rr][kk] = P[((b * NP) + (r0 + rr)) * H1 + kk] + b0[kk];
  }
  for (int i = tid; i < 16 * H1; i += 256) {
    int cc = i >> 6, kk = i & 63;
    sQ[cc][kk] = Q[((b * NP) + (c0 + cc)) * H1 + kk];
  }
  if (tid < H1) { sW0e[tid] = w0[256 * H1 + tid]; }
  else if (tid < H1 + H2) { sB1[tid - H1] = b1[tid - H1]; }
  else if (tid < H1 + 2 * H2) { sW2[tid - H1 - H2] = w2[tid - H1 - H2]; }
  if (tid < 128) {
    int rr = tid >> 4, cc = tid & 15;
    sEf[rr][cc] = efIn[((size_t)(b * NP) + (r0 + rr)) * NP + (c0 + cc)];
  }
  __syncthreads();

  const int w = tid >> 5;        // wave id -> local edge row
  const int lane = tid & 31;
  const int m = lane & 15;       // edge (col) index in tile == N col for B/C
  const int half = lane >> 4;

  const float ef = sEf[w][m];
  const float b2s = b2[0];

  v8f acc0, acc1;
  const float b1v0 = sB1[m];
  const float b1v1 = sB1[16 + m];
#pragma unroll
  for (int g = 0; g < 8; ++g) { acc0[g] = b1v0; acc1[g] = b1v1; }

#pragma unroll
  for (int kk = 0; kk < 16; ++kk) {
    const int kb = kk * 4 + half * 2;
    // A fragment: h0 built on the fly; contiguous b64 LDS loads
    const float2 pv = *(const float2*)&sPb[w][kb];
    const float2 qv = *(const float2*)&sQ[m][kb];
    const float2 wv = *(const float2*)&sW0e[kb];
    v2f a;
    a.x = relu_(fmaf(ef, wv.x, pv.x) + qv.x);
    a.y = relu_(fmaf(ef, wv.y, pv.y) + qv.y);
    // B fragments: contiguous pairs in transposed W1 -> single b64 each
    const float2 w1a = *(const float2*)&sW1T[m][kb];
    const float2 w1b = *(const float2*)&sW1T[16 + m][kb];
    v2f bb0, bb1;
    bb0.x = w1a.x; bb0.y = w1a.y;
    bb1.x = w1b.x; bb1.y = w1b.y;
    acc0 = __builtin_amdgcn_wmma_f32_16x16x4_f32(false, a, false, bb0,
                                                 (short)0, acc0, false, false);
    acc1 = __builtin_amdgcn_wmma_f32_16x16x4_f32(false, a, false, bb1,
                                                 (short)0, acc1, false, false);
  }

  // uef[m] = relu(h1) . w2 + b2 ; C layout: VGPR g holds M = g + 8*half, N = m
  const float w2a = sW2[m], w2b = sW2[16 + m];
  const int rrow = r0 + w;
#pragma unroll
  for (int g = 0; g < 8; ++g) {
    float v = fmaf(relu_(acc0[g]), w2a, relu_(acc1[g]) * w2b);
    v += __shfl_xor(v, 1, 16);
    v += __shfl_xor(v, 2, 16);
    v += __shfl_xor(v, 4, 16);
    v += __shfl_xor(v, 8, 16);
    if (m == 0) {
      int c = c0 + g + half * 8;
      uefOut[((size_t)(b * NP) + rrow) * NP + c] = v + b2s;
    }
  }
}

// ---------------------------------------------------------------------------
// aggr[b,c] = sum_r uef[b,r,c]   (column sums; all L2-resident)
// ---------------------------------------------------------------------------
__global__ void aggr_kernel(const float* __restrict__ uef,
                            float* __restrict__ aggr) {
  int t = blockIdx.x * blockDim.x + threadIdx.x;
  if (t >= NB * NN) return;
  int b = t / NN, c = t % NN;
  const float* base = uef + (size_t)b * NP * NP + c;
  float s = 0.f;
#pragma unroll 8
  for (int r = 0; r < NN; ++r) s += base[r * NP];
  aggr[b * NP + c] = s;
}

// ---------------------------------------------------------------------------
// Node MLP (129->64->32->128) + residual, in place on padded nf. Tiny.
// ---------------------------------------------------------------------------
__global__ __launch_bounds__(64) void node_mlp_kernel(
    float* __restrict__ nf, const float* __restrict__ aggr,
    const float* __restrict__ w0, const float* __restrict__ b0,
    const float* __restrict__ w1, const float* __restrict__ b1,
    const float* __restrict__ w2, const float* __restrict__ b2) {
  __shared__ float x[DD];
  __shared__ float h0[H1];
  __shared__ float h1[H2];
  int bi = blockIdx.x;            // 0 .. NB*NN-1
  int b = bi / NN, i = bi % NN;
  int node = b * NP + i;
  int t = threadIdx.x;
  x[t] = nf[node * DD + t];
  x[t + 64] = nf[node * DD + 64 + t];
  __syncthreads();
  float ag = aggr[node];
  float acc = b0[t];
#pragma unroll 8
  for (int d = 0; d < DD; ++d) acc = fmaf(x[d], w0[d * H1 + t], acc);
  acc = fmaf(ag, w0[128 * H1 + t], acc);
  h0[t] = relu_(acc);
  __syncthreads();
  if (t < H2) {
    float a2 = b1[t];
#pragma unroll 8
    for (int j = 0; j < H1; ++j) a2 = fmaf(h0[j], w1[j * H2 + t], a2);
    h1[t] = relu_(a2);
  }
  __syncthreads();
  float o0 = b2[t], o1 = b2[t + 64];
#pragma unroll 8
  for (int j = 0; j < H2; ++j) {
    float h = h1[j];
    o0 = fmaf(h, w2[j * DD + t], o0);
    o1 = fmaf(h, w2[j * DD + 64 + t], o1);
  }
  nf[node * DD + t] = x[t] + o0;
  nf[node * DD + 64 + t] = x[t + 64] + o1;
}

// ---------------------------------------------------------------------------
// Unpad nf into d_out[0 : B*N*D]
// ---------------------------------------------------------------------------
__global__ void pack_kernel(const float* __restrict__ nf,
                            float* __restrict__ out) {
  int t = blockIdx.x * blockDim.x + threadIdx.x;
  if (t >= NB * NN * DD) return;
  int d = t % DD;
  int i = (t / DD) % NN;
  int b = t / (DD * NN);
  out[t] = nf[((b * NP) + i) * DD + d];
}

// ---------------------------------------------------------------------------
extern "C" void kernel_launch(void* const* d_in, const int* in_sizes, int n_in,
                              void* d_out, int out_size, void* d_ws,
                              size_t ws_size, hipStream_t stream) {
  const float* locs   = (const float*)d_in[0];
  const float* W_init = (const float*)d_in[1];
  const float* b_init = (const float*)d_in[2];
  const float* ew0 = (const float*)d_in[3];
  const float* eb0 = (const float*)d_in[4];
  const float* ew1 = (const float*)d_in[5];
  const float* eb1 = (const float*)d_in[6];
  const float* ew2 = (const float*)d_in[7];
  const float* eb2 = (const float*)d_in[8];
  const float* nw0 = (const float*)d_in[9];
  const float* nb0 = (const float*)d_in[10];
  const float* nw1 = (const float*)d_in[11];
  const float* nb1 = (const float*)d_in[12];
  const float* nw2 = (const float*)d_in[13];
  const float* nb2 = (const float*)d_in[14];
  float* out = (float*)d_out;
  float* ws = (float*)d_ws;

  // Workspace layout (floats): ~4.5 MB total, all L2-resident on MI455X
  float* nf   = ws;
  float* Pb   = nf + (size_t)NB * NP * DD;
  float* Qb   = Pb + (size_t)NB * NP * H1;
  float* efA  = Qb + (size_t)NB * NP * H1;
  float* efB  = efA + (size_t)NB * NP * NP;
  float* aggr = efB + (size_t)NB * NP * NP;

  init_kernel<<<(NB * NP * DD + 255) / 256, 256, 0, stream>>>(
      locs, W_init, b_init, nf, out + (size_t)NB * NN * DD);
  ef_kernel<<<dim3(NP / 16, NP / 16, NB), 256, 0, stream>>>(nf, efA);

  for (int l = 0; l < NLAY; ++l) {
    const float* w0 = ew0 + (size_t)l * 257 * H1;
    const float* b0 = eb0 + (size_t)l * H1;
    const float* w1 = ew1 + (size_t)l * H1 * H2;
    const float* b1 = eb1 + (size_t)l * H2;
    const float* w2 = ew2 + (size_t)l * H2;
    const float* b2 = eb2 + l;
    float* efin  = (l & 1) ? efB : efA;
    float* efout = (l & 1) ? efA : efB;

    pq_kernel<<<NB * NP, 64, 0, stream>>>(nf, w0, Pb, Qb);
    edge_mlp_kernel<<<dim3(NP / 16, NP / 8, NB), 256, 0, stream>>>(
        Pb, Qb, efin, efout, w0, b0, w1, b1, w2, b2);
    aggr_kernel<<<(NB * NN + 255) / 256, 256, 0, stream>>>(efout, aggr);
    node_mlp_kernel<<<NB * NN, 64, 0, stream>>>(
        nf, aggr, nw0 + (size_t)l * 129 * H1, nb0 + (size_t)l * H1,
        nw1 + (size_t)l * H1 * H2, nb1 + (size_t)l * H2,
        nw2 + (size_t)l * H2 * DD, nb2 + (size_t)l * DD);
  }

  pack_kernel<<<(NB * NN * DD + 255) / 256, 256, 0, stream>>>(nf, out);
}